// KrylovSlow_57604101374536
// MI455X (gfx1250) — compile-verified
//
#include <hip/hip_runtime.h>
#include <hip/hip_bf16.h>

typedef float v2f __attribute__((ext_vector_type(2)));
typedef float v8f __attribute__((ext_vector_type(8)));

// C[64x64] = A[64x64] * B[64x64] via V_WMMA_F32_16X16X4_F32, 8 waves.
// Wave w: tile row = w>>1 (16 rows), column half = (w&1)*32 (2 tiles of 16).
// Fragment layouts per CDNA5 ISA 7.12.2:
//   A 16x4 f32 : VGPR0 = K (K=0 lanes0-15 / K=2 lanes16-31), VGPR1 = K+1
//   B 4x16 f32 : VGPR0 = row K (same half-wave split),        VGPR1 = row K+1
//   D 16x16 f32: VGPR i -> row (i + 8*(lane>=16)), col = lane&15
__device__ __forceinline__ void mm64(const float* __restrict__ A,
                                     const float* __restrict__ B,
                                     float* __restrict__ C,
                                     int lane, int wave) {
    const v8f vz = {0.f,0.f,0.f,0.f,0.f,0.f,0.f,0.f};
    v8f acc0 = vz, acc1 = vz;
    const int m    = lane & 15;
    const int kh   = (lane >> 4) << 1;       // 0 or 2
    const int trow = wave >> 1;              // 0..3
    const int row  = 16 * trow + m;
    const int cb   = (wave & 1) * 32;        // 0 or 32
    #pragma unroll 4
    for (int kk = 0; kk < 16; ++kk) {
        const int kb = 4 * kk + kh;
        v2f a;
        a.x = A[row * 64 + kb];
        a.y = A[row * 64 + kb + 1];
        v2f u, v;
        u.x = B[kb * 64 + cb + m];       u.y = B[(kb + 1) * 64 + cb + m];
        v.x = B[kb * 64 + cb + 16 + m];  v.y = B[(kb + 1) * 64 + cb + 16 + m];
        acc0 = __builtin_amdgcn_wmma_f32_16x16x4_f32(false, a, false, u, (short)0, acc0, false, false);
        acc1 = __builtin_amdgcn_wmma_f32_16x16x4_f32(false, a, false, v, (short)0, acc1, false, false);
    }
    const int r0 = 16 * trow + ((lane >> 4) << 3);
    #pragma unroll
    for (int i = 0; i < 8; ++i) {
        C[(r0 + i) * 64 + cb + m]      = acc0[i];
        C[(r0 + i) * 64 + cb + 16 + m] = acc1[i];
    }
}

__global__ __launch_bounds__(256)
void s4_krylov_kernel(const float* __restrict__ Ag,
                      const float* __restrict__ Bg,
                      const float* __restrict__ Cg,
                      const float* __restrict__ logdt,
                      float* __restrict__ out) {
    __shared__ float b0[64 * 64];
    __shared__ float b1[64 * 64];
    __shared__ float b2[64 * 64];
    __shared__ float b3[64 * 64];

    const int h    = blockIdx.x;
    const int t    = threadIdx.x;
    const int lane = t & 31;
    const int wave = t >> 5;

    const float* Ah = Ag + h * 4096;
    const float dt  = expf(logdt[h]);
    const float hdt = 0.5f * dt;

    // ---- Phase 1: b0 = Bm = I - (dt/2)A ; b1 = X0 = I + (dt/2)A ---------
    for (int i = t; i < 4096; i += 256) {
        const float m = hdt * Ah[i];
        const float e = ((i >> 6) == (i & 63)) ? 1.0f : 0.0f;
        b0[i] = e - m;
        b1[i] = e + m;
    }
    __syncthreads();

    // ---- Phase 2: Newton-Schulz S = Bm^-1: X <- X(2I - Bm X), 5 iters ----
    // Error E_{k+1} = E_k^2, E_0 = M^2 (||M|| <~ 0.6 worst case) -> ~1e-14.
    float* X  = b1;
    float* t1 = b2;
    float* t2 = b3;
    for (int it = 0; it < 5; ++it) {
        mm64(b0, X, t1, lane, wave);                 // Y = Bm * X
        __syncthreads();
        for (int i = t; i < 4096; i += 256) {        // Z = 2I - Y (in place)
            const float e2 = ((i >> 6) == (i & 63)) ? 2.0f : 0.0f;
            t1[i] = e2 - t1[i];
        }
        __syncthreads();
        mm64(X, t1, t2, lane, wave);                 // Xnew = X * Z
        __syncthreads();
        float* tmp = X; X = t2; t2 = tmp;
    }
    float* S = X;                                    // S = (I - M)^-1

    // ---- dB = dt * S * B  (kept in a register per thread t<64) ----------
    float dBv = 0.f;
    if (t < 64) {
        float s = 0.f;
        const float* Bh = Bg + h * 64;
        for (int k = 0; k < 64; ++k) s += S[t * 64 + k] * Bh[k];
        dBv = dt * s;
    }

    // ---- Phase 3: P = dA = S * (2I - Bm) --------------------------------
    for (int i = t; i < 4096; i += 256) {            // t1 = E = 2I - Bm
        const float e2 = ((i >> 6) == (i & 63)) ? 2.0f : 0.0f;
        t1[i] = e2 - b0[i];
    }
    __syncthreads();
    mm64(S, t1, t2, lane, wave);                     // P = S * E
    __syncthreads();

    // Buffer roles from here: U = b0, P/T ping-pong = {t2, t1}, W = S's buf
    float* U    = b0;
    float* Pa   = t2;
    float* Tb   = t1;
    float* Wbuf = S;

    // ---- Phase 4: U-doubling: col r of U = dA^r dB ----------------------
    for (int i = t; i < 4096; i += 256) U[i] = 0.f;
    __syncthreads();
    if (t < 64) U[t * 64 + 0] = dBv;                 // column 0 = dB
    __syncthreads();
    int wdt = 1;
    for (int j = 0; j < 6; ++j) {
        mm64(Pa, U, Tb, lane, wave);                 // T = dA^(2^j) * U
        __syncthreads();
        for (int i = t; i < 64 * wdt; i += 256) {    // cols [0,w) -> [w,2w)
            const int r = i / wdt, c = i - r * wdt;
            U[r * 64 + wdt + c] = Tb[r * 64 + c];
        }
        __syncthreads();
        mm64(Pa, Pa, Tb, lane, wave);                // power <- power^2
        __syncthreads();
        float* tmp = Pa; Pa = Tb; Tb = tmp;
        wdt <<= 1;
    }
    // Pa = dA^64 = G; U complete.

    // ---- Phase 5: W-doubling: row q of W = C^T G^q ----------------------
    float* W = Wbuf;
    for (int i = t; i < 4096; i += 256) W[i] = 0.f;
    __syncthreads();
    if (t < 64) W[t] = Cg[h * 64 + t];               // row 0 = C^T
    __syncthreads();
    wdt = 1;
    for (int j = 0; j < 6; ++j) {
        mm64(W, Pa, Tb, lane, wave);                 // T = W * G^(2^j)
        __syncthreads();
        for (int i = t; i < wdt * 64; i += 256) {    // rows [0,w) -> [w,2w)
            W[(wdt + (i >> 6)) * 64 + (i & 63)] = Tb[i];
        }
        __syncthreads();
        if (j < 5) {
            mm64(Pa, Pa, Tb, lane, wave);            // G <- G^2
            __syncthreads();
            float* tmp = Pa; Pa = Tb; Tb = tmp;
        }
        wdt <<= 1;
    }

    // ---- Phase 6: OUT = W * U -> global: out[h, 64q + r] ----------------
    mm64(W, U, out + h * 4096, lane, wave);
}

extern "C" void kernel_launch(void* const* d_in, const int* in_sizes, int n_in,
                              void* d_out, int out_size, void* d_ws, size_t ws_size,
                              hipStream_t stream) {
    const float* A     = (const float*)d_in[0];   // (H,64,64) f32
    const float* B     = (const float*)d_in[1];   // (H,64)    f32
    const float* C     = (const float*)d_in[2];   // (H,64)    f32
    const float* logdt = (const float*)d_in[3];   // (H,)      f32
    (void)d_ws; (void)ws_size; (void)n_in; (void)out_size;

    const int H = in_sizes[3];                    // 256 heads
    s4_krylov_kernel<<<dim3(H), dim3(256), 0, stream>>>(
        A, B, C, logdt, (float*)d_out);
}